// PraxisMemory_74113955660085
// MI455X (gfx1250) — compile-verified
//
#include <hip/hip_runtime.h>
#include <hip/hip_bf16.h>

// ---- CDNA5 (gfx1250) f32 WMMA helpers ------------------------------------
typedef float v2f __attribute__((ext_vector_type(2)));
typedef float v8f __attribute__((ext_vector_type(8)));

#define WMMA_F32(a, b, c) \
  __builtin_amdgcn_wmma_f32_16x16x4_f32(false, (a), false, (b), (short)0, (c), false, false)

// Problem constants
#define NB 2
#define NS 256
#define ND 512
#define NV 32000
#define SP1 257        // S+1
#define NROWS 512      // B*S
#define XPB 192        // (K_SIM+K_CONT)*MEM_LEN rows per batch
#define XROWS 384      // B*XPB
#define CTX 448        // XPB + S
#define VSPLIT 10      // vocab split for k_lse_part
#define VT_PER_SPLIT 200   // (NV/16)/VSPLIT tiles per split

// d_out layout (floats)
#define OUT_Q 0
#define OUT_C1 262144
#define OUT_C2 720896
#define OUT_M 1179648

// ---------------------------------------------------------------------------
// K1a: fused partial GEMM (q @ brain_w^T) + flash (max, sum-exp) over a vocab
// split. Grid = 32 row-tiles x VSPLIT. 4 waves/block, 2 N-tiles per wave per
// K-pass sharing one A fragment (2 independent WMMA chains).
__global__ __launch_bounds__(128) void k_lse_part(const float* __restrict__ q,
                                                  const float* __restrict__ W,
                                                  float* __restrict__ pmx,
                                                  float* __restrict__ psm) {
  __shared__ __align__(16) float As[16 * 516];   // padded stride (bank-safe)
  __shared__ float Pmx[4][32][8];
  __shared__ float Psm[4][32][8];
  const int mt = blockIdx.x / VSPLIT, vs = blockIdx.x % VSPLIT;
  const int m0 = mt * 16;
  // Stage A tile into LDS with gfx1250 async copy (per-lane 16B, GV mode).
  {
    const int tid = threadIdx.x;
    for (int j = 0; j < 16; ++j) {
      unsigned lo = (unsigned)(uintptr_t)(&As[j * 516 + tid * 4]);
      const float* gp = q + (size_t)(m0 + j) * 512 + tid * 4;
      asm volatile("global_load_async_to_lds_b128 %0, %1, off"
                   :: "v"(lo), "v"(gp) : "memory");
    }
    asm volatile("s_wait_asynccnt 0x0" ::: "memory");
  }
  __syncthreads();
  const int wv = threadIdx.x >> 5, lane = threadIdx.x & 31;
  const int hh = lane >> 4, nn = lane & 15, kk = hh * 2;
  float mx[8], sm[8];
  for (int r = 0; r < 8; ++r) { mx[r] = -3.0e38f; sm[r] = 0.f; }
  const float* arow = &As[nn * 516 + kk];
  for (int it = 0; it < VT_PER_SPLIT / 8; ++it) {
    const int vt = vs * VT_PER_SPLIT + it * 8 + wv * 2;   // 2 adjacent tiles
    v8f acc0 = {}, acc1 = {};
    const float* w0 = W + (size_t)(vt * 16 + nn) * 512 + kk;
    const float* w1 = w0 + (size_t)16 * 512;
    for (int k0 = 0; k0 < 512; k0 += 4) {
      v2f a = *(const v2f*)&arow[k0];
      v2f b0 = *(const v2f*)&w0[k0];
      v2f b1 = *(const v2f*)&w1[k0];
      acc0 = WMMA_F32(a, b0, acc0);
      acc1 = WMMA_F32(a, b1, acc1);
    }
    for (int r = 0; r < 8; ++r) {     // online softmax (column-set agnostic)
      float x = acc0[r];
      if (x > mx[r]) { sm[r] = sm[r] * expf(mx[r] - x) + 1.f; mx[r] = x; }
      else           { sm[r] += expf(x - mx[r]); }
      x = acc1[r];
      if (x > mx[r]) { sm[r] = sm[r] * expf(mx[r] - x) + 1.f; mx[r] = x; }
      else           { sm[r] += expf(x - mx[r]); }
    }
  }
  for (int r = 0; r < 8; ++r) { Pmx[wv][lane][r] = mx[r]; Psm[wv][lane][r] = sm[r]; }
  __syncthreads();
  if (threadIdx.x < 16) {             // deterministic block combine
    int m = threadIdx.x;
    int base = (m >= 8) ? 16 : 0, r = m & 7;
    float M = -3.0e38f;
    for (int w = 0; w < 4; ++w)
      for (int l = 0; l < 16; ++l) M = fmaxf(M, Pmx[w][base + l][r]);
    float Ssum = 0.f;
    for (int w = 0; w < 4; ++w)
      for (int l = 0; l < 16; ++l)
        Ssum += Psm[w][base + l][r] * expf(Pmx[w][base + l][r] - M);
    pmx[(size_t)(m0 + m) * VSPLIT + vs] = M;
    psm[(size_t)(m0 + m) * VSPLIT + vs] = Ssum;
  }
}

// K1b: combine vocab-split partials -> logsumexp per row
__global__ __launch_bounds__(256) void k_lse_comb(const float* __restrict__ pmx,
                                                  const float* __restrict__ psm,
                                                  float* __restrict__ lse) {
  int r = blockIdx.x * 256 + threadIdx.x;   // 512 rows
  float M = -3.0e38f;
  for (int v = 0; v < VSPLIT; ++v) M = fmaxf(M, pmx[(size_t)r * VSPLIT + v]);
  float S = 0.f;
  for (int v = 0; v < VSPLIT; ++v)
    S += psm[(size_t)r * VSPLIT + v] * expf(pmx[(size_t)r * VSPLIT + v] - M);
  lse[r] = M + logf(S);
}

// K2: surprise = lse - <q, W[target]>
__global__ __launch_bounds__(128) void k_surprise(const float* __restrict__ q,
                                                  const float* __restrict__ W,
                                                  const int* __restrict__ tgt,
                                                  const float* __restrict__ lse,
                                                  float* __restrict__ surp) {
  __shared__ float part[128];
  int r = blockIdx.x;
  const float* qr = q + (size_t)r * 512;
  const float* wr = W + (size_t)tgt[r] * 512;
  float s = 0.f;
  for (int k = threadIdx.x; k < 512; k += 128) s += qr[k] * wr[k];
  part[threadIdx.x] = s;
  __syncthreads();
  if (threadIdx.x == 0) {
    float acc = 0.f;
    for (int i = 0; i < 128; ++i) acc += part[i];
    surp[r] = lse[r] - acc;
  }
}

// K3: sliding-window threshold (ddof=1) -> boundaries
__global__ __launch_bounds__(256) void k_bound(const float* __restrict__ surp,
                                               float* __restrict__ bnd) {
  __shared__ float s[256];
  __shared__ float thrw[237];
  int b = blockIdx.x;
  s[threadIdx.x] = surp[b * 256 + threadIdx.x];
  __syncthreads();
  int w = threadIdx.x;
  if (w < 237) {
    float mean = 0.f;
    for (int i = 0; i < 20; ++i) mean += s[w + i];
    mean *= (1.0f / 20.0f);
    float var = 0.f;
    for (int i = 0; i < 20; ++i) { float d = s[w + i] - mean; var += d * d; }
    var *= (1.0f / 19.0f);
    thrw[w] = mean + 2.0f * sqrtf(var);
  }
  __syncthreads();
  int t = threadIdx.x;
  float thr = (t < 19) ? thrw[0] : thrw[t - 19];
  bnd[b * 256 + t] = (s[t] > thr) ? 1.f : 0.f;
}

// K4: sims[b] = q_b q_b^T  (WMMA, 1 wave per 16x16 tile)
__global__ __launch_bounds__(32) void k_sims(const float* __restrict__ q,
                                             float* __restrict__ sims) {
  int blk = blockIdx.x;
  int b = blk >> 8, mt = (blk >> 4) & 15, nt = blk & 15;
  int lane = threadIdx.x;
  int hh = lane >> 4, nn = lane & 15, kk = hh * 2;
  const float* Abase = q + ((size_t)(b * 256 + mt * 16 + nn)) * 512 + kk;
  const float* Bbase = q + ((size_t)(b * 256 + nt * 16 + nn)) * 512 + kk;
  v8f acc = {};
  for (int k0 = 0; k0 < 512; k0 += 4) {
    v2f a = *(const v2f*)&Abase[k0];
    v2f bb = *(const v2f*)&Bbase[k0];
    acc = WMMA_F32(a, bb, acc);
  }
  for (int r = 0; r < 8; ++r) {
    int m = mt * 16 + r + 8 * hh;
    int n = nt * 16 + nn;
    sims[((size_t)b * 256 + m) * 256 + n] = acc[r];
  }
}

// K5: degrees + total per batch (deterministic)
__global__ __launch_bounds__(256) void k_deg(const float* __restrict__ sims,
                                             float* __restrict__ deg,
                                             float* __restrict__ tot) {
  __shared__ float ds[256];
  int b = blockIdx.x, i = threadIdx.x;
  const float* row = sims + ((size_t)b * 256 + i) * 256;
  float s = 0.f;
  for (int j = 0; j < 256; ++j) s += row[j];
  ds[i] = s;
  deg[b * 256 + i] = s;
  __syncthreads();
  if (i == 0) {
    float T = 0.f;
    for (int j = 0; j < 256; ++j) T += ds[j];
    tot[b] = T;
  }
}

// K6: modularity+conductance for candidate flips (cand==256 -> no flip).
__global__ __launch_bounds__(256) void k_metrics(const float* __restrict__ sims,
                                                 const float* __restrict__ bnd,
                                                 const float* __restrict__ deg,
                                                 const float* __restrict__ tot,
                                                 float* __restrict__ newQ,
                                                 float* __restrict__ newC) {
  __shared__ float bs[256];
  __shared__ int com[256];
  __shared__ float ds[256];
  __shared__ float rowsum[256];
  __shared__ float Dc[257];
  __shared__ float Sc[257];
  int b = blockIdx.x / SP1, cand = blockIdx.x % SP1;
  int tid = threadIdx.x;
  float bv = bnd[b * 256 + tid];
  if (tid == cand) bv = 0.f;
  bs[tid] = bv;
  ds[tid] = deg[b * 256 + tid];
  __syncthreads();
  if (tid == 0) {
    int run = 0;
    for (int i = 0; i < 256; ++i) { run += (int)bs[i]; com[i] = run; }
  }
  __syncthreads();
  int ci = com[tid];
  const float* row = sims + ((size_t)b * 256 + tid) * 256;
  float s = 0.f;
  for (int j = 0; j < 256; ++j) if (com[j] == ci) s += row[j];
  rowsum[tid] = s;
  __syncthreads();
  for (int c = tid; c < 257; c += 256) {
    float Dv = 0.f, Sv = 0.f;
    for (int i = 0; i < 256; ++i)
      if (com[i] == c) { Dv += ds[i]; Sv += rowsum[i]; }
    Dc[c] = Dv; Sc[c] = Sv;
  }
  __syncthreads();
  if (tid == 0) {
    float T = tot[b];
    float te = T + 1e-10f;
    float qsum = 0.f, csum = 0.f;
    for (int c = 0; c < 257; ++c) {
      qsum += Sc[c] - Dc[c] * Dc[c] / te;
      float minv = fminf(Dc[c], T - Dc[c]);
      csum += (Dc[c] - Sc[c]) / (minv + 1e-10f);
    }
    newQ[b * SP1 + cand] = qsum / te;
    newC[b * SP1 + cand] = csum / ((float)com[255] + 1.0f);
  }
}

// K7: refine boundaries, labels, segment (start,count) tables
__global__ __launch_bounds__(256) void k_segments(const float* __restrict__ bnd,
                                                  const float* __restrict__ newQ,
                                                  const float* __restrict__ newC,
                                                  int* __restrict__ seg_start,
                                                  int* __restrict__ seg_count) {
  int b = blockIdx.x, tid = threadIdx.x;
  seg_count[b * SP1 + tid] = 0;
  seg_start[b * SP1 + tid] = 0;
  if (tid == 0) { seg_count[b * SP1 + 256] = 0; seg_start[b * SP1 + 256] = 0; }
  __syncthreads();
  if (tid == 0) {
    float cQ = newQ[b * SP1 + 256], cC = newC[b * SP1 + 256];
    int lab = 0, prev = -1;
    for (int t = 0; t < 256; ++t) {
      float bv = bnd[b * 256 + t];
      float r = bv;
      if (bv > 0.f) {
        bool impr = (newQ[b * SP1 + t] > cQ) || (newC[b * SP1 + t] < cC);
        if (impr) r = 0.f;
      }
      lab += (int)r;
      if (lab != prev) { seg_start[b * SP1 + lab] = t; prev = lab; }
      seg_count[b * SP1 + lab] += 1;
    }
  }
}

// K8: flat rank -> memory slot assignment (last write wins, as .set scatter)
__global__ __launch_bounds__(256) void k_slots(const int* __restrict__ seg_start,
                                               const int* __restrict__ seg_count,
                                               const int* __restrict__ meml_in,
                                               int* __restrict__ sw, int* __restrict__ sb,
                                               int* __restrict__ sst, int* __restrict__ sct,
                                               int* __restrict__ slen) {
  int tid = threadIdx.x;
  sw[tid] = 0;
  __syncthreads();
  if (tid == 0) {
    int rank = 0;
    for (int f = 0; f < NB * SP1; ++f) {
      int ct = seg_count[f];
      if (ct > 0) {
        int slot = rank & 255;
        sw[slot] = 1;
        sb[slot] = f / SP1;
        sst[slot] = seg_start[f];
        sct[slot] = ct;
        ++rank;
      }
    }
    for (int s = 0; s < 256; ++s)
      slen[s] = sw[s] ? (sct[s] < 16 ? sct[s] : 16) : meml_in[s];
  }
}

// K9: updated memory keys (segment head query row, or original key)
__global__ __launch_bounds__(256) void k_keys(const float* __restrict__ q,
                                              const float* __restrict__ memk_in,
                                              const int* __restrict__ sw,
                                              const int* __restrict__ sb,
                                              const int* __restrict__ sst,
                                              float* __restrict__ keys) {
  int s = blockIdx.x;
  const float* src = sw[s] ? (q + ((size_t)(sb[s] * 256 + sst[s])) * 512)
                           : (memk_in + (size_t)s * 512);
  for (int d = threadIdx.x; d < 512; d += 256) keys[(size_t)s * 512 + d] = src[d];
}

// K10a: qmean[b,d] = mean_s q[b,s,d]
__global__ __launch_bounds__(256) void k_qmean(const float* __restrict__ q,
                                               float* __restrict__ qmean) {
  int idx = blockIdx.x * 256 + threadIdx.x;   // 1024
  int b = idx >> 9, d = idx & 511;
  float s = 0.f;
  for (int t = 0; t < 256; ++t) s += q[((size_t)(b * 256 + t)) * 512 + d];
  qmean[idx] = s * (1.0f / 256.0f);
}

// K10b: qm = qmean @ sim_w^T + sim_b
__global__ __launch_bounds__(256) void k_qm(const float* __restrict__ qmean,
                                            const float* __restrict__ simw,
                                            const float* __restrict__ simb,
                                            float* __restrict__ qm) {
  int idx = blockIdx.x * 256 + threadIdx.x;   // 1024
  int b = idx >> 9, e = idx & 511;
  const float* wr = simw + (size_t)e * 512;
  const float* qr = qmean + (size_t)b * 512;
  float s = 0.f;
  for (int d = 0; d < 512; ++d) s += qr[d] * wr[d];
  qm[idx] = s + simb[e];
}

// K11: scores = qm @ keys^T with invalid masking
__global__ __launch_bounds__(256) void k_scores(const float* __restrict__ qm,
                                                const float* __restrict__ keys,
                                                const int* __restrict__ slen,
                                                float* __restrict__ scores) {
  int idx = blockIdx.x * 256 + threadIdx.x;   // 512
  int b = idx >> 8, s = idx & 255;
  const float* kr = keys + (size_t)s * 512;
  const float* qr = qm + (size_t)b * 512;
  float v = 0.f;
  for (int d = 0; d < 512; ++d) v += qr[d] * kr[d];
  scores[idx] = (slen[s] > 0) ? v : -1e9f;
}

// K12: stable top-k (first-index wins ties, like lax.top_k)
__global__ void k_topk(const float* __restrict__ scores,
                       const int* __restrict__ slen,
                       int* __restrict__ topi, int* __restrict__ conti) {
  if (threadIdx.x == 0 && blockIdx.x == 0) {
    for (int b = 0; b < NB; ++b) {
      bool used[256];
      for (int i = 0; i < 256; ++i) used[i] = false;
      for (int k = 0; k < 8; ++k) {
        float best = -3.4e38f; int bi = 0;
        for (int s = 0; s < 256; ++s) {
          if (used[s]) continue;
          float v = scores[b * 256 + s];
          if (v > best) { best = v; bi = s; }
        }
        used[bi] = true;
        topi[b * 8 + k] = bi;
      }
    }
    bool used[256];
    for (int i = 0; i < 256; ++i) used[i] = false;
    for (int k = 0; k < 4; ++k) {
      int best = -2147483647, bi = 0;
      for (int s = 0; s < 256; ++s) {
        if (used[s]) continue;
        int v = (slen[s] > 0) ? s : -1;
        if (v > best) { best = v; bi = s; }
      }
      used[bi] = true;
      conti[k] = bi;
    }
  }
}

// K13: gather memory rows into X (B, 192, 512): [sim_buf | cont_buf]
__global__ __launch_bounds__(256) void k_buildX(const float* __restrict__ q,
                                                const float* __restrict__ memv_in,
                                                const int* __restrict__ sw,
                                                const int* __restrict__ sb,
                                                const int* __restrict__ sst,
                                                const int* __restrict__ sct,
                                                const int* __restrict__ topi,
                                                const int* __restrict__ conti,
                                                float* __restrict__ X) {
  int blk = blockIdx.x;                 // B*192
  int b = blk / XPB, rr = blk % XPB;
  int p = rr & 15;
  int slot = (rr < 128) ? topi[b * 8 + (rr >> 4)] : conti[(rr - 128) >> 4];
  float* xr = X + (size_t)blk * 512;
  if (sw[slot]) {
    int cnt = sct[slot];
    int lim = cnt < 16 ? cnt : 16;
    if (p < lim) {
      const float* src = q + ((size_t)(sb[slot] * 256 + sst[slot] + p)) * 512;
      for (int d = threadIdx.x; d < 512; d += 256) xr[d] = src[d];
    } else {
      for (int d = threadIdx.x; d < 512; d += 256) xr[d] = 0.f;
    }
  } else {
    const float* src = memv_in + ((size_t)slot * 16 + p) * 512;
    for (int d = threadIdx.x; d < 512; d += 256) xr[d] = src[d];
  }
}

// K14/K15: Out = [relu](A @ Wt^T + bias); 4 waves/block, 1 n-tile each (WMMA)
__global__ __launch_bounds__(128) void k_gemm(const float* __restrict__ A,
                                              const float* __restrict__ Wt,
                                              const float* __restrict__ bias,
                                              float* __restrict__ Out, int relu) {
  int blk = blockIdx.x;                 // (M/16) * (512/64)
  int mt = blk / 8, nt4 = blk % 8;
  int wv = threadIdx.x >> 5, lane = threadIdx.x & 31;
  int nt = nt4 * 4 + wv;
  int hh = lane >> 4, nn = lane & 15, kk = hh * 2;
  const float* Abase = A + ((size_t)(mt * 16 + nn)) * 512 + kk;
  const float* Bbase = Wt + ((size_t)(nt * 16 + nn)) * 512 + kk;
  v8f acc = {};
  for (int k0 = 0; k0 < 512; k0 += 4) {
    v2f a = *(const v2f*)&Abase[k0];
    v2f bb = *(const v2f*)&Bbase[k0];
    acc = WMMA_F32(a, bb, acc);
  }
  for (int r = 0; r < 8; ++r) {
    int m = mt * 16 + r + 8 * hh;
    int n = nt * 16 + nn;
    float v = acc[r] + bias[n];
    if (relu) v = fmaxf(v, 0.f);
    Out[(size_t)m * 512 + n] = v;
  }
}

// K16: assemble outputs
__global__ void k_out_query(const float* __restrict__ q, float* __restrict__ out) {
  size_t i = (size_t)blockIdx.x * 256 + threadIdx.x;   // 262144
  out[OUT_Q + i] = q[i];
}
__global__ void k_out_ctx(const float* __restrict__ Y, const float* __restrict__ key,
                          float* __restrict__ out) {
  size_t i = (size_t)blockIdx.x * 256 + threadIdx.x;   // B*448*512 = 458752
  int d = (int)(i & 511);
  size_t row = i >> 9;
  int b = (int)(row / CTX), r = (int)(row % CTX);
  float v = (r < XPB) ? Y[((size_t)(b * XPB + r)) * 512 + d]
                      : key[((size_t)(b * 256 + (r - XPB))) * 512 + d];
  out[OUT_C1 + i] = v;
  out[OUT_C2 + i] = v;
}
__global__ void k_out_mask(const float* __restrict__ am, float* __restrict__ out) {
  int i = blockIdx.x * 256 + threadIdx.x;
  if (i < NB * CTX) {
    int b = i / CTX, c = i % CTX;
    out[OUT_M + i] = (c < XPB) ? 1.0f : am[b * 256 + (c - XPB)];
  }
}

// ---------------------------------------------------------------------------
extern "C" void kernel_launch(void* const* d_in, const int* in_sizes, int n_in,
                              void* d_out, int out_size, void* d_ws, size_t ws_size,
                              hipStream_t stream) {
  (void)in_sizes; (void)n_in; (void)out_size; (void)ws_size;
  const float* q       = (const float*)d_in[0];
  const float* key     = (const float*)d_in[1];
  const float* amask   = (const float*)d_in[3];
  const int*   tgt     = (const int*)d_in[4];
  const float* brain_w = (const float*)d_in[5];
  const float* w1      = (const float*)d_in[6];
  const float* b1      = (const float*)d_in[7];
  const float* w2      = (const float*)d_in[8];
  const float* b2      = (const float*)d_in[9];
  const float* simw    = (const float*)d_in[10];
  const float* simb    = (const float*)d_in[11];
  const float* memk_in = (const float*)d_in[12];
  const float* memv_in = (const float*)d_in[13];
  const int*   meml_in = (const int*)d_in[14];

  float* wsf = (float*)d_ws;
  size_t o = 0;
  auto A = [&](size_t n) { size_t r = o; o += (n + 15) & ~(size_t)15; return r; };
  float* pmx       = wsf + A((size_t)NROWS * VSPLIT);
  float* psm       = wsf + A((size_t)NROWS * VSPLIT);
  float* lse       = wsf + A(512);
  float* surp      = wsf + A(512);
  float* bnd       = wsf + A(512);
  float* sims      = wsf + A((size_t)NB * 256 * 256);
  float* deg       = wsf + A(512);
  float* tot       = wsf + A(16);
  float* newQ      = wsf + A(NB * SP1);
  float* newC      = wsf + A(NB * SP1);
  int*   seg_start = (int*)(wsf + A(NB * SP1));
  int*   seg_count = (int*)(wsf + A(NB * SP1));
  int*   sw        = (int*)(wsf + A(256));
  int*   sb        = (int*)(wsf + A(256));
  int*   sst       = (int*)(wsf + A(256));
  int*   sct       = (int*)(wsf + A(256));
  int*   slen      = (int*)(wsf + A(256));
  float* keys      = wsf + A((size_t)256 * 512);
  float* qmean     = wsf + A(1024);
  float* qm        = wsf + A(1024);
  float* scores    = wsf + A(512);
  int*   topi      = (int*)(wsf + A(16));
  int*   conti     = (int*)(wsf + A(16));
  float* X         = wsf + A((size_t)XROWS * 512);
  float* H         = wsf + A((size_t)XROWS * 512);
  float* Y         = wsf + A((size_t)XROWS * 512);

  k_lse_part<<<(NROWS / 16) * VSPLIT, 128, 0, stream>>>(q, brain_w, pmx, psm);
  k_lse_comb<<<2, 256, 0, stream>>>(pmx, psm, lse);
  k_surprise<<<NROWS, 128, 0, stream>>>(q, brain_w, tgt, lse, surp);
  k_bound<<<NB, 256, 0, stream>>>(surp, bnd);
  k_sims<<<NB * 16 * 16, 32, 0, stream>>>(q, sims);
  k_deg<<<NB, 256, 0, stream>>>(sims, deg, tot);
  k_metrics<<<NB * SP1, 256, 0, stream>>>(sims, bnd, deg, tot, newQ, newC);
  k_segments<<<NB, 256, 0, stream>>>(bnd, newQ, newC, seg_start, seg_count);
  k_slots<<<1, 256, 0, stream>>>(seg_start, seg_count, meml_in, sw, sb, sst, sct, slen);
  k_keys<<<256, 256, 0, stream>>>(q, memk_in, sw, sb, sst, keys);
  k_qmean<<<4, 256, 0, stream>>>(q, qmean);
  k_qm<<<4, 256, 0, stream>>>(qmean, simw, simb, qm);
  k_scores<<<2, 256, 0, stream>>>(qm, keys, slen, scores);
  k_topk<<<1, 32, 0, stream>>>(scores, slen, topi, conti);
  k_buildX<<<NB * XPB, 256, 0, stream>>>(q, memv_in, sw, sb, sst, sct, topi, conti, X);
  k_gemm<<<(XROWS / 16) * 8, 128, 0, stream>>>(X, w1, b1, H, 1);
  k_gemm<<<(XROWS / 16) * 8, 128, 0, stream>>>(H, w2, b2, Y, 0);
  k_out_query<<<262144 / 256, 256, 0, stream>>>(q, (float*)d_out);
  k_out_ctx<<<458752 / 256, 256, 0, stream>>>(Y, key, (float*)d_out);
  k_out_mask<<<4, 256, 0, stream>>>(amask, (float*)d_out);
}